// Commnet_87660282512012
// MI455X (gfx1250) — compile-verified
//
#include <hip/hip_runtime.h>

typedef __attribute__((ext_vector_type(2))) float v2f;
typedef __attribute__((ext_vector_type(8))) float v8f;

#define BN 64     // agents per batch element
#define DD 128    // feature dim
#define OO 16     // output dim
#define RS 132    // padded LDS row stride for H2 (conflict-free WMMA A fetch)

__global__ __launch_bounds__(128) void commnet_fused(
    const float* __restrict__ obs,
    const float* __restrict__ wH1,
    const float* __restrict__ wH2,
    const float* __restrict__ wC2,
    const float* __restrict__ wout,
    const float* __restrict__ bout,
    float* __restrict__ out)
{
    __shared__ float sObs[BN * DD];   // obs tile (skip connection)
    __shared__ float sH[BN * RS];     // H1 then H2, padded rows
    __shared__ float sW[DD * OO];     // w_out staged
    __shared__ float sB[OO];          // b_out staged

    const int t = threadIdx.x;              // 0..127
    const float* obsB = obs + (size_t)blockIdx.x * (BN * DD);

    // Stage head weights into LDS (2048 + 16 floats).
    #pragma unroll
    for (int i = 0; i < (DD * OO) / 128; ++i)
        sW[t + i * 128] = wout[t + i * 128];
    if (t < OO) sB[t] = bout[t];

    // Thread t owns feature column d = t for all 64 agents of this batch elem.
    const int d = t;
    const float a1 = wH1[d];
    const float a2 = wH2[d];
    const float c2 = wC2[d];

    // Phase 1a: load obs (coalesced), compute H1, column-local sum over agents.
    float s1 = 0.f;
    #pragma unroll 8
    for (int n = 0; n < BN; ++n) {
        float o = obsB[n * DD + d];
        sObs[n * DD + d] = o;
        float h = tanhf(a1 * o);
        sH[n * RS + d] = h;
        s1 += h;
    }

    // Phase 1b: C1 = (sum - H1)/(N-1); H2 = tanh(wH2*H1 + wC2*C1) + obs.
    const float inv = 1.f / (float)(BN - 1);
    #pragma unroll 8
    for (int n = 0; n < BN; ++n) {
        float h1 = sH[n * RS + d];
        float o  = sObs[n * DD + d];
        float c1 = (s1 - h1) * inv;
        float h2 = tanhf(a2 * h1 + c2 * c1) + o;
        sH[n * RS + d] = h2;
    }

    __syncthreads();

    // Phase 2: actions = H2 @ w_out + b_out via V_WMMA_F32_16X16X4_F32.
    // 4 waves, wave w handles agents [w*16, w*16+16). K = 128 -> 32 WMMAs.
    const int lane = t & 31;
    const int wv   = t >> 5;
    const int m0   = wv * 16;
    const int row  = lane & 15;   // A: M index
    const int half = lane >> 4;   // A/B: selects K pair (0 -> K,K+1 ; 1 -> K+2,K+3)
    const int col  = lane & 15;   // B/D: N index

    v8f acc = {};
    #pragma unroll
    for (int kk = 0; kk < 32; ++kk) {
        const int k = kk * 4 + half * 2;
        v2f av, bv;
        av.x = sH[(m0 + row) * RS + k];
        av.y = sH[(m0 + row) * RS + k + 1];
        bv.x = sW[k * OO + col];
        bv.y = sW[(k + 1) * OO + col];
        acc = __builtin_amdgcn_wmma_f32_16x16x4_f32(
            /*neg_a=*/false, av, /*neg_b=*/false, bv,
            /*c_mod=*/(short)0, acc, /*reuse_a=*/false, /*reuse_b=*/false);
    }

    // D layout: VGPR r -> M = r (lanes 0-15) / r+8 (lanes 16-31), N = lane%16.
    const float bb = sB[col];
    float* outB = out + (size_t)blockIdx.x * (BN * OO);
    #pragma unroll
    for (int r = 0; r < 8; ++r) {
        const int m = m0 + r + half * 8;
        outB[m * OO + col] = acc[r] + bb;
    }
}

extern "C" void kernel_launch(void* const* d_in, const int* in_sizes, int n_in,
                              void* d_out, int out_size, void* d_ws, size_t ws_size,
                              hipStream_t stream) {
    // setup_inputs order: obs, w_H1, w_C1 (unused: multiplies zero C0),
    //                     w_H2, w_C2, w_out, b_out
    const float* obs  = (const float*)d_in[0];
    const float* wH1  = (const float*)d_in[1];
    const float* wH2  = (const float*)d_in[3];
    const float* wC2  = (const float*)d_in[4];
    const float* wout = (const float*)d_in[5];
    const float* bout = (const float*)d_in[6];
    float* out = (float*)d_out;

    const int B = in_sizes[0] / (BN * DD);   // 8192
    commnet_fused<<<B, 128, 0, stream>>>(obs, wH1, wH2, wC2, wout, bout, out);
}